// SSIM_28063316312584
// MI455X (gfx1250) — compile-verified
//
#include <hip/hip_runtime.h>
#include <math.h>

// ---------------- tile geometry ----------------
#define TH 32            // output tile rows
#define TW 32            // output tile cols
#define IH 42            // TH + 10 (halo rows)
#define IW 43            // TW + 11 (halo cols + 1 extra col)
#define THP 44           // tmp rows incl. 2 zero pad rows (max rel row = 16+27 = 43)
#define TMS 33           // LDS stride for horizontal-conv output (odd -> no bank conflicts)
#define NT 128           // 4 wave32 per block, one 16x16 subtile each

typedef __attribute__((ext_vector_type(2))) float v2f;
typedef __attribute__((ext_vector_type(8))) float v8f;

__device__ __forceinline__ int reflect512(int i) {
    if (i < 0) i = -i;
    if (i > 511) i = 1022 - i;
    return i;
}

// ---------------- ws init (ws is poisoned by harness) ----------------
__global__ void init_ws_kernel(unsigned int* umax, double* dsum) {
    *umax = 0u;
    *dsum = 0.0;
}

// ---------------- pass 1: max over gray(img1) for data_range ----------------
__global__ __launch_bounds__(256) void max_gray_kernel(const float* __restrict__ img1,
                                                       unsigned int* __restrict__ umax) {
    const int gid    = blockIdx.x * blockDim.x + threadIdx.x;
    const int stride = gridDim.x * blockDim.x;
    float m = 0.0f;
    for (int pix = gid; pix < 32 * 512 * 512; pix += stride) {
        int n  = pix >> 18;          // image index
        int rc = pix & 262143;       // row*512+col
        int p  = n * 786432 + rc;    // < 2^25, fits int
        float g = 0.299f * img1[p] + 0.587f * img1[p + 262144] + 0.114f * img1[p + 524288];
        m = fmaxf(m, g);
    }
    for (int off = 16; off > 0; off >>= 1)
        m = fmaxf(m, __shfl_down(m, off, 32));
    __shared__ float red[8];
    int lane = threadIdx.x & 31, wave = threadIdx.x >> 5;
    if (lane == 0) red[wave] = m;
    __syncthreads();
    if (threadIdx.x == 0) {
        float mm = red[0];
        #pragma unroll
        for (int i = 1; i < 8; ++i) mm = fmaxf(mm, red[i]);
        atomicMax(umax, __float_as_uint(mm));   // gray >= 0 -> bit order == value order
    }
}

// ---------------- pass 2: fused gray + separable 11x11 Gaussian + SSIM ----------------
__global__ __launch_bounds__(NT) void ssim_kernel(const float* __restrict__ img1,
                                                  const float* __restrict__ img2,
                                                  const unsigned int* __restrict__ umax,
                                                  double* __restrict__ dsum) {
    __shared__ float g1s[IH * IW];
    __shared__ float g2s[IH * IW];
    __shared__ float tmp[5][THP * TMS];  // horizontal-conv results, 5 channels (+2 zero rows)
    __shared__ float swc[11];            // 1D Gaussian weights (for dynamic indexing)
    __shared__ double redd[4];

    const int tid = threadIdx.x;
    const int t   = blockIdx.x & 255;    // 16x16 tiles per image
    const int img = blockIdx.x >> 8;
    const int r0  = (t >> 4) * TH;
    const int c0  = (t & 15) * TW;

    // --- Gaussian 1D weights (window is separable; w = e / sum(e)) ---
    float w[11];
    {
        float s = 0.0f;
        #pragma unroll
        for (int i = 0; i < 11; ++i) {
            float d = (float)(i - 5);
            w[i] = expf(-d * d * (1.0f / 4.5f));   // 2*sigma^2 = 4.5
            s += w[i];
        }
        float inv = 1.0f / s;
        #pragma unroll
        for (int i = 0; i < 11; ++i) w[i] *= inv;
        if (tid == 0) {
            #pragma unroll
            for (int i = 0; i < 11; ++i) swc[i] = w[i];
        }
    }

    const float dr = (__uint_as_float(*umax) <= 1.1f) ? 1.0f : 255.0f;
    const float C1 = (0.01f * dr) * (0.01f * dr);
    const float C2 = (0.03f * dr) * (0.03f * dr);

    // --- phase 1: reflect-padded gray load of halo region ---
    const int ibase = img * 786432;
    for (int idx = tid; idx < IH * IW; idx += NT) {
        int r  = idx / IW, c = idx % IW;
        int gr = reflect512(r0 - 5 + r);
        int gc = reflect512(c0 - 5 + c);
        int p  = ibase + gr * 512 + gc;
        float a0 = img1[p], a1 = img1[p + 262144], a2 = img1[p + 524288];
        float b0 = img2[p], b1 = img2[p + 262144], b2 = img2[p + 524288];
        g1s[idx] = 0.299f * a0 + 0.587f * a1 + 0.114f * a2;
        g2s[idx] = 0.299f * b0 + 0.587f * b1 + 0.114f * b2;
    }
    __syncthreads();

    // --- phase 2: horizontal 11-tap conv of the 5 channels ---
    for (int idx = tid; idx < IH * TW; idx += NT) {
        int r = idx / TW, c = idx % TW;
        const float* p1 = &g1s[r * IW + c];
        const float* p2 = &g2s[r * IW + c];
        float s0 = 0, s1 = 0, s2 = 0, s3 = 0, s4 = 0;
        #pragma unroll
        for (int j = 0; j < 11; ++j) {
            float a = p1[j], b = p2[j], wj = w[j];
            s0 = fmaf(wj, a, s0);
            s1 = fmaf(wj, b, s1);
            s2 = fmaf(wj * a, a, s2);
            s3 = fmaf(wj * b, b, s3);
            s4 = fmaf(wj * a, b, s4);
        }
        int o = r * TMS + c;
        tmp[0][o] = s0; tmp[1][o] = s1; tmp[2][o] = s2; tmp[3][o] = s3; tmp[4][o] = s4;
    }
    // zero the 2 pad rows (42,43) so phase-3 B loads are unconditional & finite
    if (tid < 2 * TW) {
        int o = (IH + (tid >> 5)) * TMS + (tid & 31);
        tmp[0][o] = 0.0f; tmp[1][o] = 0.0f; tmp[2][o] = 0.0f; tmp[3][o] = 0.0f; tmp[4][o] = 0.0f;
    }
    __syncthreads();

    // --- phase 3: vertical 11-tap conv as banded matmul on WMMA f32 16x16x4 ---
    // out[i][c] = sum_m A[i][m] * tmp[sr*16+m][col],  A[i][m] = w[m-i] (banded, m in 0..25;
    // m=26,27 have A==0, so reading real data or the zero pad rows there is harmless)
    const int lane = tid & 31;
    const int wave = tid >> 5;
    const int sr   = wave >> 1;          // subtile row (0..1)
    const int sc   = wave & 1;           // subtile col (0..1)
    const int li   = lane & 15;
    const int hi   = lane >> 4;          // 0 or 1 (upper half-wave)

    // A fragments per K-chunk: VGPR0 holds K = 4kc + 2*hi, VGPR1 holds K+1 (ISA A-layout)
    float ax[7], ay[7];
    #pragma unroll
    for (int kc = 0; kc < 7; ++kc) {
        int kb = 4 * kc + 2 * hi;
        int d0 = kb - li;
        int d1 = kb + 1 - li;
        ax[kc] = ((unsigned)d0 < 11u) ? swc[d0] : 0.0f;
        ay[kc] = ((unsigned)d1 < 11u) ? swc[d1] : 0.0f;
    }

    const int col = sc * 16 + li;
    const int rbase = sr * 16;
    float U[5][8];
    #pragma unroll
    for (int ch = 0; ch < 5; ++ch) {
        v8f acc = {0.f, 0.f, 0.f, 0.f, 0.f, 0.f, 0.f, 0.f};
        #pragma unroll
        for (int kc = 0; kc < 7; ++kc) {
            int mb = 4 * kc + 2 * hi;                 // B row for VGPR0 (K ordering 0,2 / 1,3)
            v2f b;
            b.x = tmp[ch][(rbase + mb    ) * TMS + col];
            b.y = tmp[ch][(rbase + mb + 1) * TMS + col];
            v2f a; a.x = ax[kc]; a.y = ay[kc];
            acc = __builtin_amdgcn_wmma_f32_16x16x4_f32(false, a, false, b,
                                                        (short)0, acc, false, false);
        }
        #pragma unroll
        for (int v = 0; v < 8; ++v) U[ch][v] = acc[v];
    }

    // --- phase 4: SSIM map on C-fragment registers (VGPR v: M=v / M=v+8) ---
    const float covn = 121.0f / 120.0f;   // WIN*WIN/(WIN*WIN-1)
    float lsum = 0.0f;
    #pragma unroll
    for (int v = 0; v < 8; ++v) {
        int row  = r0 + sr * 16 + v + 8 * hi;
        int gcol = c0 + sc * 16 + li;
        float ux = U[0][v], uy = U[1][v], uxx = U[2][v], uyy = U[3][v], uxy = U[4][v];
        float vx  = covn * (uxx - ux * ux);
        float vy  = covn * (uyy - uy * uy);
        float vxy = covn * (uxy - ux * uy);
        float A1 = 2.0f * ux * uy + C1;
        float A2 = 2.0f * vxy + C2;
        float B1 = ux * ux + uy * uy + C1;
        float B2 = vx + vy + C2;
        float S  = (A1 * A2) / (B1 * B2);
        if (row >= 5 && row <= 506 && gcol >= 5 && gcol <= 506)
            lsum += S;
    }

    // --- phase 5: reduce and accumulate ---
    for (int off = 16; off > 0; off >>= 1)
        lsum += __shfl_down(lsum, off, 32);
    if (lane == 0) redd[wave] = (double)lsum;
    __syncthreads();
    if (tid == 0) {
        double s = redd[0] + redd[1] + redd[2] + redd[3];
        atomicAdd(dsum, s);
    }
}

// ---------------- pass 3: mean ----------------
__global__ void finalize_kernel(const double* __restrict__ dsum, float* __restrict__ out) {
    out[0] = (float)(*dsum / 8064128.0);   // 32 * 502 * 502
}

extern "C" void kernel_launch(void* const* d_in, const int* in_sizes, int n_in,
                              void* d_out, int out_size, void* d_ws, size_t ws_size,
                              hipStream_t stream) {
    const float* img1 = (const float*)d_in[0];
    const float* img2 = (const float*)d_in[1];
    unsigned int* umax = (unsigned int*)d_ws;
    double* dsum = (double*)((char*)d_ws + 8);

    init_ws_kernel<<<1, 1, 0, stream>>>(umax, dsum);
    max_gray_kernel<<<1024, 256, 0, stream>>>(img1, umax);
    ssim_kernel<<<32 * 256, NT, 0, stream>>>(img1, img2, umax, dsum);
    finalize_kernel<<<1, 1, 0, stream>>>(dsum, (float*)d_out);
}